// Model_23733989278167
// MI455X (gfx1250) — compile-verified
//
#include <hip/hip_runtime.h>
#include <hip/hip_bf16.h>

// ---------------------------------------------------------------------------
// Deformable-conv model for MI455X (gfx1250, wave32).
//   Kernel 1 (prep):   offset = conv3x3(x,w0,b0); mask = sigmoid(conv3x3(x,w2,b2))
//                      -> workspace (L2-resident: x+offset+mask = 161MB < 192MB L2)
//   Kernel 2 (deform): shared bilinear sampling for x1/x2, then two f32 WMMA
//                      GEMM chains D[16oc x 16px] += W[16x4] * S[4x16], 27 steps.
// ---------------------------------------------------------------------------

typedef __attribute__((ext_vector_type(2))) float v2f;
typedef __attribute__((ext_vector_type(8))) float v8f;

#define NB   16
#define CIN  12
#define HH   256
#define WW   256
#define KK   9
#define CK   108          // CIN*KK
#define HO   254
#define WO   254
#define COUT 16
#define HWO  (HO * WO)    // 64516
#define TX   16           // 16-pixel tiles per row (ceil(254/16))

// ------------------------------ prep kernel --------------------------------
__global__ __launch_bounds__(256) void prep_kernel(
    const float* __restrict__ x,
    const float* __restrict__ w0, const float* __restrict__ b0,
    const float* __restrict__ w2, const float* __restrict__ b2,
    float* __restrict__ offs, float* __restrict__ msk)
{
    int p = blockIdx.x * 256 + threadIdx.x;   // pixel within one image
    int b = blockIdx.y;
    if (p >= HWO) return;
    int y  = p / WO;
    int xo = p % WO;

    float acc[27];
#pragma unroll
    for (int i = 0; i < 27; ++i) acc[i] = 0.f;

    const float* xb = x + (size_t)b * CIN * HH * WW;
    for (int c = 0; c < CIN; ++c) {
        const float* xc = xb + (size_t)c * HH * WW + (size_t)y * WW + xo;
        float tap[9];
#pragma unroll
        for (int i = 0; i < 3; ++i)
#pragma unroll
            for (int j = 0; j < 3; ++j)
                tap[i * 3 + j] = xc[i * WW + j];
#pragma unroll
        for (int t = 0; t < 9; ++t) {
            float v = tap[t];
#pragma unroll
            for (int oc = 0; oc < 18; ++oc)
                acc[oc] = fmaf(v, w0[oc * CK + c * 9 + t], acc[oc]);   // uniform -> SMEM
#pragma unroll
            for (int oc = 0; oc < 9; ++oc)
                acc[18 + oc] = fmaf(v, w2[oc * CK + c * 9 + t], acc[18 + oc]);
        }
    }

    size_t ob = (size_t)b * 18 * HWO + p;
#pragma unroll
    for (int oc = 0; oc < 18; ++oc)
        offs[ob + (size_t)oc * HWO] = acc[oc] + b0[oc];

    size_t mb = (size_t)b * 9 * HWO + p;
#pragma unroll
    for (int oc = 0; oc < 9; ++oc) {
        float v = acc[18 + oc] + b2[oc];
        msk[mb + (size_t)oc * HWO] = 1.f / (1.f + __expf(-v));
    }
}

// ----------------------------- deform kernel -------------------------------
// 4 waves / block; each wave owns one 16-pixel tile (b, y, x0..x0+15) and
// produces out1/out2 [16 oc x 16 px] via V_WMMA_F32_16X16X4_F32 chains.
__global__ __launch_bounds__(128) void deform_kernel(
    const float* __restrict__ x,
    const float* __restrict__ w1, const float* __restrict__ b1,
    const float* __restrict__ w3, const float* __restrict__ b3,
    const float* __restrict__ offs, const float* __restrict__ msk,
    float* __restrict__ out)
{
    __shared__ __align__(16) float wA[2][COUT * CK];     // A-matrices (weights)
    __shared__ __align__(16) float biasL[2][16];
    __shared__ __align__(16) float accL[4][16 * CK];     // raw samples   [px][ck]
    __shared__ __align__(16) float accM[4][16 * CK];     // masked samples[px][ck]

    const int tid  = threadIdx.x;
    const int lane = tid & 31;
    const int wv   = tid >> 5;

    for (int i = tid; i < COUT * CK; i += 128) {
        wA[0][i] = w1[i];
        wA[1][i] = w3[i];
    }
    if (tid < 16) { biasL[0][tid] = b1[tid]; biasL[1][tid] = b3[tid]; }
    __syncthreads();

    // tile decomposition: 65024 tiles total, exactly 4 per block
    int tile = blockIdx.x * 4 + wv;
    int b    = tile / (HO * TX);
    int rem  = tile % (HO * TX);
    int y    = rem / TX;
    int x0   = (rem % TX) * 16;

    const float* xb = x + (size_t)b * CIN * HH * WW;

    // ---- shared bilinear sampling: 9 taps x 16 pixels per tile ----
    for (int idx = lane; idx < KK * 16; idx += 32) {
        int k  = idx >> 4;            // 0..8
        int px = idx & 15;            // 0..15
        int xo = x0 + px; if (xo > WO - 1) xo = WO - 1;   // clamp partial tile
        int ky = k / 3, kx = k % 3;

        size_t obase = ((size_t)b * 18 + 2 * k) * HWO + (size_t)y * WO + xo;
        float dy = offs[obase];
        float dx = offs[obase + HWO];
        float m  = msk[((size_t)b * 9 + k) * HWO + (size_t)y * WO + xo];

        float py  = dy + (float)(ky + y);
        float pxf = dx + (float)(kx + xo);
        float y0f = floorf(py), x0f = floorf(pxf);
        float fy = py - y0f, fx = pxf - x0f;
        int y0i = (int)y0f, x0i = (int)x0f;

        bool vy0 = (y0i >= 0) & (y0i < HH);
        bool vy1 = (y0i + 1 >= 0) & (y0i + 1 < HH);
        bool vx0 = (x0i >= 0) & (x0i < WW);
        bool vx1 = (x0i + 1 >= 0) & (x0i + 1 < WW);
        float w00 = (1.f - fy) * (1.f - fx) * (float)(vy0 && vx0);
        float w01 = (1.f - fy) * fx         * (float)(vy0 && vx1);
        float w10 = fy * (1.f - fx)         * (float)(vy1 && vx0);
        float w11 = fy * fx                 * (float)(vy1 && vx1);

        int yc0 = min(max(y0i, 0), HH - 1), yc1 = min(max(y0i + 1, 0), HH - 1);
        int xc0 = min(max(x0i, 0), WW - 1), xc1 = min(max(x0i + 1, 0), WW - 1);
        int i00 = yc0 * WW + xc0, i01 = yc0 * WW + xc1;
        int i10 = yc1 * WW + xc0, i11 = yc1 * WW + xc1;

        float* aL = &accL[wv][px * CK + k];
        float* aM = &accM[wv][px * CK + k];
#pragma unroll
        for (int c = 0; c < CIN; ++c) {
            const float* xc = xb + (size_t)c * HH * WW;   // whole x is L2-resident
            float v = w00 * xc[i00] + w01 * xc[i01] + w10 * xc[i10] + w11 * xc[i11];
            aL[c * 9] = v;
            aM[c * 9] = v * m;
        }
    }
    __syncthreads();

    // ---- WMMA phase: D = W * S (+bias), 27 steps of K=4, two chains ----
    const int half = lane >> 4;     // 0: K{0,1}, 1: K{2,3}
    const int l16  = lane & 15;     // A: M row / B: N column
    const int koff = half * 2;

    v8f c1, c2;
#pragma unroll
    for (int r = 0; r < 8; ++r) {
        c1[r] = biasL[0][r + half * 8];
        c2[r] = biasL[1][r + half * 8];
    }

    const float* wa1 = &wA[0][l16 * CK + koff];
    const float* wa3 = &wA[1][l16 * CK + koff];
    const float* bl  = &accL[wv][l16 * CK + koff];
    const float* bm  = &accM[wv][l16 * CK + koff];

#pragma unroll
    for (int s = 0; s < 27; ++s) {
        v2f a1 = *(const v2f*)(wa1 + 4 * s);   // even offsets -> 8B aligned ds_load_b64
        v2f bu = *(const v2f*)(bl + 4 * s);
        c1 = __builtin_amdgcn_wmma_f32_16x16x4_f32(false, a1, false, bu,
                                                   (short)0, c1, false, false);
        v2f a2 = *(const v2f*)(wa3 + 4 * s);
        v2f bk = *(const v2f*)(bm + 4 * s);
        c2 = __builtin_amdgcn_wmma_f32_16x16x4_f32(false, a2, false, bk,
                                                   (short)0, c2, false, false);
    }

    // ---- store: VGPR r holds (oc = r + 8*half, px = l16) ----
    int px = l16;
    if (x0 + px < WO) {
        size_t pixo = (size_t)y * WO + x0 + px;
        size_t x2off = (size_t)NB * COUT * HWO;
#pragma unroll
        for (int r = 0; r < 8; ++r) {
            int oc = r + half * 8;
            size_t o = ((size_t)b * COUT + oc) * HWO + pixo;
            out[o]         = c1[r];
            out[o + x2off] = c2[r];
        }
    }
}

// ------------------------------- launcher ----------------------------------
extern "C" void kernel_launch(void* const* d_in, const int* in_sizes, int n_in,
                              void* d_out, int out_size, void* d_ws, size_t ws_size,
                              hipStream_t stream) {
    const float* x  = (const float*)d_in[0];
    const float* w0 = (const float*)d_in[1];
    const float* b0 = (const float*)d_in[2];
    const float* w1 = (const float*)d_in[3];
    const float* b1 = (const float*)d_in[4];
    const float* w2 = (const float*)d_in[5];
    const float* b2 = (const float*)d_in[6];
    const float* w3 = (const float*)d_in[7];
    const float* b3 = (const float*)d_in[8];
    float* out  = (float*)d_out;
    float* offs = (float*)d_ws;                         // [16,18,254,254]
    float* msk  = offs + (size_t)NB * 18 * HWO;         // [16, 9,254,254]

    dim3 gA((HWO + 255) / 256, NB);
    prep_kernel<<<gA, 256, 0, stream>>>(x, w0, b0, w2, b2, offs, msk);

    int tiles = NB * HO * TX;                           // 65024, exactly /4
    deform_kernel<<<tiles / 4, 128, 0, stream>>>(x, w1, b1, w3, b3,
                                                 offs, msk, out);
}